// ComENet_82652350644686
// MI455X (gfx1250) — compile-verified
//
#include <hip/hip_runtime.h>
#include <hip/hip_bf16.h>

// ---------------------------------------------------------------------------
// ComENet-like GNN block, fp32, CDNA5 (gfx1250), wave32.
// Heavy GEMMs via V_WMMA_F32_16X16X4_F32 (exact fp32; workload is HBM-bound on
// streaming feature1 at ~321 MB, so fp32 matrix throughput is not the limiter).
// All hot loops are straight-line (no EXEC divergence around WMMA) and
// double-buffered so global loads pipeline past the matrix ops.
// ---------------------------------------------------------------------------

#define NNODES  12800
#define NEDGES  51200
#define HID     256
#define F1DIM   1568
#define F2DIM   224
#define NGRAPH  64
#define NLAYERS 3
#define EPSV    1e-5f

typedef __attribute__((ext_vector_type(2))) float v2f;
typedef __attribute__((ext_vector_type(8))) float v8f;

__device__ __forceinline__ float swish_f(float v) {
    return v / (1.0f + __expf(-v));
}

// ---------------------------------------------------------------------------
// Node GEMM, Nout fixed = HID = 256.
//   out[M,256] = op( A[M,K] @ W^T + bias ) (+ addsrc),  W row-major [256, K].
// Block = 128 threads = 4 waves; wave w owns columns [w*64, w*64+64).
// Grid = (M/16, 1). No guards anywhere: EXEC stays all-ones through WMMA.
// K multiple of 4. Double-buffered K loop.
// ---------------------------------------------------------------------------
__global__ void gemm_node_wmma(const float* __restrict__ A, int lda,
                               const float* __restrict__ W, int ldw,
                               const float* __restrict__ bias,
                               const float* __restrict__ addsrc,
                               float* __restrict__ out,
                               int K, int accum, int act)
{
    const int lane = threadIdx.x & 31;
    const int wave = threadIdx.x >> 5;
    const int half = lane >> 4;     // 0: lanes 0-15, 1: lanes 16-31
    const int ml   = lane & 15;     // row (A) / col (B,C) index within tile

    const int row0    = blockIdx.x * 16;
    const int colbase = wave * 64;

    v8f acc[4];
    if (accum) {
#pragma unroll
        for (int t = 0; t < 4; ++t)
#pragma unroll
            for (int r = 0; r < 8; ++r)
                acc[t][r] = out[(size_t)(row0 + r + 8 * half) * HID
                                + colbase + t * 16 + ml];
    } else {
#pragma unroll
        for (int t = 0; t < 4; ++t)
            acc[t] = (v8f){0.f,0.f,0.f,0.f,0.f,0.f,0.f,0.f};
    }

    // A fragment (ISA 7.12.2, 32-bit A 16x4): lane m = lane&15,
    // v0 <- K=k+2*half, v1 <- K=k+2*half+1  (contiguous pair -> b64 load).
    const float* arow = A + (size_t)(row0 + ml) * lda + 2 * half;
    // B fragment mirrors A in K: rows of W (W^T columns), contiguous pair.
    const float* wrow[4];
#pragma unroll
    for (int t = 0; t < 4; ++t)
        wrow[t] = W + (size_t)(colbase + t * 16 + ml) * ldw + 2 * half;

    // ---- software-pipelined K loop (double buffer, last iteration peeled) ----
    v2f a_cur, b_cur[4];
    a_cur.x = arow[0]; a_cur.y = arow[1];
#pragma unroll
    for (int t = 0; t < 4; ++t) { b_cur[t].x = wrow[t][0]; b_cur[t].y = wrow[t][1]; }

#pragma unroll 2
    for (int k = 0; k + 4 < K; k += 4) {
        v2f a_nxt, b_nxt[4];
        a_nxt.x = arow[k + 4]; a_nxt.y = arow[k + 5];
#pragma unroll
        for (int t = 0; t < 4; ++t) {
            b_nxt[t].x = wrow[t][k + 4];
            b_nxt[t].y = wrow[t][k + 5];
        }
#pragma unroll
        for (int t = 0; t < 4; ++t)
            acc[t] = __builtin_amdgcn_wmma_f32_16x16x4_f32(
                false, a_cur, false, b_cur[t], (short)0, acc[t], false, false);
        a_cur = a_nxt;
#pragma unroll
        for (int t = 0; t < 4; ++t) b_cur[t] = b_nxt[t];
    }
#pragma unroll
    for (int t = 0; t < 4; ++t)
        acc[t] = __builtin_amdgcn_wmma_f32_16x16x4_f32(
            false, a_cur, false, b_cur[t], (short)0, acc[t], false, false);

    // ---- epilogue ----
#pragma unroll
    for (int t = 0; t < 4; ++t) {
        const int n  = colbase + t * 16 + ml;
        const float bb = bias ? bias[n] : 0.0f;
#pragma unroll
        for (int r = 0; r < 8; ++r) {
            const int m = row0 + r + 8 * half;
            float v = acc[t][r] + bb;
            if (act) v = swish_f(v);
            if (addsrc) v += addsrc[(size_t)m * HID + n];
            out[(size_t)m * HID + n] = v;
        }
    }
}

// ---------------------------------------------------------------------------
// Weight-collapse GEMM:  out[M,Nout] = A[M,K] @ W  (W row-major [K, Nout]).
// One wave per block (blockDim = 32), one 16x16 tile per wave.
// Grid = (M/16, Nout/16) exactly (1568 and 224 are multiples of 16) -> no guards.
// Tiny GEMM (<= 103 M MACs), simplicity over tiling.
// ---------------------------------------------------------------------------
__global__ void gemm_wcollapse(const float* __restrict__ A, int lda,
                               const float* __restrict__ W, int ldw,
                               float* __restrict__ out, int ldo, int K)
{
    const int lane = threadIdx.x & 31;
    const int half = lane >> 4;
    const int ml   = lane & 15;

    const int row0 = blockIdx.x * 16;
    const int col0 = blockIdx.y * 16;

    v8f acc = (v8f){0.f,0.f,0.f,0.f,0.f,0.f,0.f,0.f};

    const float* arow = A + (size_t)(row0 + ml) * lda + 2 * half;
    const float* wcol = W + (size_t)(2 * half) * ldw + col0 + ml;

#pragma unroll 2
    for (int k = 0; k < K; k += 4) {
        v2f a, b;
        a.x = arow[k];
        a.y = arow[k + 1];
        b.x = wcol[(size_t)k * ldw];
        b.y = wcol[(size_t)(k + 1) * ldw];
        acc = __builtin_amdgcn_wmma_f32_16x16x4_f32(
            false, a, false, b, (short)0, acc, false, false);
    }

#pragma unroll
    for (int r = 0; r < 8; ++r)
        out[(size_t)(row0 + r + 8 * half) * ldo + col0 + ml] = acc[r];
}

// ---------------------------------------------------------------------------
// Fused edge kernel:  f[e,:] = feature[e,:] @ W12^T  (W12 row-major [HID,Kf]),
// then msg = f[e,:] * x1[j[e],:], scattered with global_atomic_add_f32 into
// agg[i[e],:].  Wave: 16 edges x 64 channels; block (4 waves) covers all 256
// channels so the streamed feature tile is read once per block (L0/L2 reuse).
// ---------------------------------------------------------------------------
__global__ void edge_gemm_scatter(const float* __restrict__ Fe,
                                  const float* __restrict__ W12,
                                  const float* __restrict__ x1,
                                  const int*   __restrict__ eidx,
                                  float* __restrict__ agg,
                                  int E, int Kf)
{
    const int lane = threadIdx.x & 31;
    const int wave = threadIdx.x >> 5;
    const int half = lane >> 4;
    const int ml   = lane & 15;

    const int row0    = blockIdx.x * 16;     // edge tile
    const int colbase = wave * 64;           // channel strip

    v8f acc[4];
#pragma unroll
    for (int t = 0; t < 4; ++t) acc[t] = (v8f){0.f,0.f,0.f,0.f,0.f,0.f,0.f,0.f};

    const float* arow = Fe + (size_t)(row0 + ml) * Kf + 2 * half;
    const float* wrow[4];
#pragma unroll
    for (int t = 0; t < 4; ++t)
        wrow[t] = W12 + (size_t)(colbase + t * 16 + ml) * Kf + 2 * half;

    v2f a_cur, b_cur[4];
    a_cur.x = arow[0]; a_cur.y = arow[1];
#pragma unroll
    for (int t = 0; t < 4; ++t) { b_cur[t].x = wrow[t][0]; b_cur[t].y = wrow[t][1]; }

#pragma unroll 2
    for (int k = 0; k + 4 < Kf; k += 4) {
        v2f a_nxt, b_nxt[4];
        a_nxt.x = arow[k + 4]; a_nxt.y = arow[k + 5];
#pragma unroll
        for (int t = 0; t < 4; ++t) {
            b_nxt[t].x = wrow[t][k + 4];
            b_nxt[t].y = wrow[t][k + 5];
        }
#pragma unroll
        for (int t = 0; t < 4; ++t)
            acc[t] = __builtin_amdgcn_wmma_f32_16x16x4_f32(
                false, a_cur, false, b_cur[t], (short)0, acc[t], false, false);
        a_cur = a_nxt;
#pragma unroll
        for (int t = 0; t < 4; ++t) b_cur[t] = b_nxt[t];
    }
#pragma unroll
    for (int t = 0; t < 4; ++t)
        acc[t] = __builtin_amdgcn_wmma_f32_16x16x4_f32(
            false, a_cur, false, b_cur[t], (short)0, acc[t], false, false);

    // epilogue: gather x1[j], scatter-add into agg[i] (atomics land in L2)
#pragma unroll
    for (int r = 0; r < 8; ++r) {
        const int e  = row0 + r + 8 * half;
        const int jn = eidx[e];          // source node (row 0 of edge_index)
        const int in = eidx[E + e];      // target node (row 1)
        const float* xr = x1  + (size_t)jn * HID;
        float*       ar = agg + (size_t)in * HID;
#pragma unroll
        for (int t = 0; t < 4; ++t) {
            const int n = colbase + t * 16 + ml;
            atomicAdd(&ar[n], acc[t][r] * xr[n]);
        }
    }
}

// ---------------------------------------------------------------------------
// Small elementwise / reduction kernels
// ---------------------------------------------------------------------------
__global__ void zero_f32(float* __restrict__ p, int n)
{
    const int i = blockIdx.x * blockDim.x + threadIdx.x;
    if (i < n) p[i] = 0.0f;
}

__global__ void graph_stats_sum(const float* __restrict__ h,
                                const int*   __restrict__ batch,
                                float* __restrict__ sum,
                                float* __restrict__ cnt)
{
    const int idx = blockIdx.x * blockDim.x + threadIdx.x;
    if (idx >= NNODES * HID) return;
    const int node = idx >> 8;
    const int c    = idx & (HID - 1);
    const int g    = batch[node];
    atomicAdd(&sum[g * HID + c], h[idx]);
    if (c == 0) atomicAdd(&cnt[g], 1.0f);
}

__global__ void graph_mean(float* __restrict__ sum, const float* __restrict__ cnt)
{
    const int idx = blockIdx.x * blockDim.x + threadIdx.x;
    if (idx >= NGRAPH * HID) return;
    float c = cnt[idx >> 8];
    c = (c < 1.0f) ? 1.0f : c;
    sum[idx] = sum[idx] / c;     // sum -> mean in place
}

__global__ void graph_center_var(const float* __restrict__ h,
                                 const int*   __restrict__ batch,
                                 const float* __restrict__ mean,
                                 const float* __restrict__ alpha,
                                 float* __restrict__ cen,
                                 float* __restrict__ var)
{
    const int idx = blockIdx.x * blockDim.x + threadIdx.x;
    if (idx >= NNODES * HID) return;
    const int node = idx >> 8;
    const int c    = idx & (HID - 1);
    const int g    = batch[node];
    const float v  = h[idx] - alpha[c] * mean[g * HID + c];
    cen[idx] = v;
    atomicAdd(&var[g * HID + c], v * v);
}

__global__ void graph_norm(float* __restrict__ cen,
                           const int*   __restrict__ batch,
                           const float* __restrict__ var,
                           const float* __restrict__ cnt,
                           const float* __restrict__ gamma,
                           const float* __restrict__ beta)
{
    const int idx = blockIdx.x * blockDim.x + threadIdx.x;
    if (idx >= NNODES * HID) return;
    const int node = idx >> 8;
    const int c    = idx & (HID - 1);
    const int g    = batch[node];
    float ct = cnt[g];
    ct = (ct < 1.0f) ? 1.0f : ct;
    const float istd = rsqrtf(var[g * HID + c] / ct + EPSV);
    cen[idx] = gamma[c] * cen[idx] * istd + beta[c];
}

// ---------------------------------------------------------------------------
// Orchestration
// ---------------------------------------------------------------------------
extern "C" void kernel_launch(void* const* d_in, const int* in_sizes, int n_in,
                              void* d_out, int out_size, void* d_ws, size_t ws_size,
                              hipStream_t stream)
{
    // inputs in setup_inputs() dict order
    const float* x        = (const float*)d_in[0];
    const float* feature1 = (const float*)d_in[1];
    const float* feature2 = (const float*)d_in[2];
    const int*   eidx     = (const int*)  d_in[3];
    const int*   batch    = (const int*)  d_in[4];
    const float* lin_w    = (const float*)d_in[5];
    const float* lin_b    = (const float*)d_in[6];
    const float* f1_w1    = (const float*)d_in[7];
    const float* f1_w2    = (const float*)d_in[8];
    const float* f2_w1    = (const float*)d_in[9];
    const float* f2_w2    = (const float*)d_in[10];
    const float* c1_ll_w  = (const float*)d_in[11];
    const float* c1_ll_b  = (const float*)d_in[12];
    const float* c1_lr_w  = (const float*)d_in[13];
    const float* c2_ll_w  = (const float*)d_in[14];
    const float* c2_ll_b  = (const float*)d_in[15];
    const float* c2_lr_w  = (const float*)d_in[16];
    const float* lin1_w   = (const float*)d_in[17];
    const float* lin1_b   = (const float*)d_in[18];
    const float* lin2_w   = (const float*)d_in[19];
    const float* lin2_b   = (const float*)d_in[20];
    const float* lincat_w = (const float*)d_in[21];
    const float* lincat_b = (const float*)d_in[22];
    const float* lins_w   = (const float*)d_in[23];
    const float* lins_b   = (const float*)d_in[24];
    const float* ngamma   = (const float*)d_in[25];
    const float* nbeta    = (const float*)d_in[26];
    const float* nalpha   = (const float*)d_in[27];
    const float* final_w  = (const float*)d_in[28];
    const float* final_b  = (const float*)d_in[29];
    float*       out      = (float*)d_out;

    // workspace layout (floats)
    const size_t NH = (size_t)NNODES * HID;          // 3,276,800
    float* ws   = (float*)d_ws;
    float* x1   = ws;                                 // swish(lin(x))
    float* agg1 = ws + NH;                            // edge agg 1 / later h1
    float* agg2 = ws + 2 * NH;                        // edge agg 2 / later h2
    float* t1   = ws + 3 * NH;                        // scratch A
    float* t2   = ws + 4 * NH;                        // scratch B
    float* w12a = ws + 5 * NH;                        // [HID, F1DIM]
    float* w12b = w12a + (size_t)HID * F1DIM;         // [HID, F2DIM]
    float* ssum = w12b + (size_t)HID * F2DIM;         // [NGRAPH, HID] sum -> mean
    float* svar = ssum + (size_t)NGRAPH * HID;        // [NGRAPH, HID]
    float* scnt = svar + (size_t)NGRAPH * HID;        // [NGRAPH]

    const dim3 blk(128);                              // 4 waves (wave32)
    const dim3 gNode(NNODES / 16, 1);
    const dim3 gEdge(NEDGES / 16, 1);
    const int  EW = 256;                              // elementwise block

    // 0) zero atomic targets: agg1+agg2 (contiguous), stats (contiguous)
    {
        const int nAgg = (int)(2 * NH);
        zero_f32<<<(nAgg + EW - 1) / EW, EW, 0, stream>>>(agg1, nAgg);
        const int nSt = 2 * NGRAPH * HID + NGRAPH;
        zero_f32<<<(nSt + EW - 1) / EW, EW, 0, stream>>>(ssum, nSt);
    }

    // 1) collapse edge-MLP weight pairs (no activation between them):
    //    W12 = w2 @ w1, so feature @ w1^T @ w2^T == feature @ W12^T
    gemm_wcollapse<<<dim3(HID / 16, F1DIM / 16), dim3(32), 0, stream>>>(
        f1_w2, HID, f1_w1, F1DIM, w12a, F1DIM, HID);
    gemm_wcollapse<<<dim3(HID / 16, F2DIM / 16), dim3(32), 0, stream>>>(
        f2_w2, HID, f2_w1, F2DIM, w12b, F2DIM, HID);

    // 2) entry projection: x1 = swish(x @ lin_w^T + lin_b)
    gemm_node_wmma<<<gNode, blk, 0, stream>>>(x, HID, lin_w, HID, lin_b, nullptr,
                                              x1, HID, 0, 1);

    // 3) edge branches: f = feature @ W12^T; agg[i] += f * x1[j]
    edge_gemm_scatter<<<gEdge, blk, 0, stream>>>(feature1, w12a, x1, eidx, agg1,
                                                 NEDGES, F1DIM);
    edge_gemm_scatter<<<gEdge, blk, 0, stream>>>(feature2, w12b, x1, eidx, agg2,
                                                 NEDGES, F2DIM);

    // 4) branch 1: t1 = agg1@c1_ll^T + b + x1@c1_lr^T ; h1 = swish(t1@lin1^T + b)
    gemm_node_wmma<<<gNode, blk, 0, stream>>>(agg1, HID, c1_ll_w, HID, c1_ll_b,
                                              nullptr, t1, HID, 0, 0);
    gemm_node_wmma<<<gNode, blk, 0, stream>>>(x1, HID, c1_lr_w, HID, nullptr,
                                              nullptr, t1, HID, 1, 0);
    gemm_node_wmma<<<gNode, blk, 0, stream>>>(t1, HID, lin1_w, HID, lin1_b,
                                              nullptr, agg1, HID, 0, 1);

    // 5) branch 2
    gemm_node_wmma<<<gNode, blk, 0, stream>>>(agg2, HID, c2_ll_w, HID, c2_ll_b,
                                              nullptr, t2, HID, 0, 0);
    gemm_node_wmma<<<gNode, blk, 0, stream>>>(x1, HID, c2_lr_w, HID, nullptr,
                                              nullptr, t2, HID, 1, 0);
    gemm_node_wmma<<<gNode, blk, 0, stream>>>(t2, HID, lin2_w, HID, lin2_b,
                                              nullptr, agg2, HID, 0, 1);

    // 6) concat linear + residual: t1 = [h1,h2]@lincat^T + b + x1
    gemm_node_wmma<<<gNode, blk, 0, stream>>>(agg1, HID, lincat_w, 2 * HID,
                                              lincat_b, nullptr, t1, HID, 0, 0);
    gemm_node_wmma<<<gNode, blk, 0, stream>>>(agg2, HID, lincat_w + HID, 2 * HID,
                                              nullptr, x1, t1, HID, 1, 0);

    // 7) residual swish layers (ping-pong t1 <-> t2): h = swish(h@W^T+b) + h
    gemm_node_wmma<<<gNode, blk, 0, stream>>>(t1, HID, lins_w + 0 * HID * HID, HID,
                                              lins_b + 0 * HID, t1, t2, HID, 0, 1);
    gemm_node_wmma<<<gNode, blk, 0, stream>>>(t2, HID, lins_w + 1 * HID * HID, HID,
                                              lins_b + 1 * HID, t2, t1, HID, 0, 1);
    gemm_node_wmma<<<gNode, blk, 0, stream>>>(t1, HID, lins_w + 2 * HID * HID, HID,
                                              lins_b + 2 * HID, t1, t2, HID, 0, 1);
    // final pre-norm h lives in t2

    // 8) GraphNorm over per-graph segments
    {
        const int nNH = NNODES * HID;
        graph_stats_sum<<<(nNH + EW - 1) / EW, EW, 0, stream>>>(t2, batch, ssum, scnt);
        graph_mean<<<(NGRAPH * HID + EW - 1) / EW, EW, 0, stream>>>(ssum, scnt);
        graph_center_var<<<(nNH + EW - 1) / EW, EW, 0, stream>>>(t2, batch, ssum,
                                                                 nalpha, agg1, svar);
        graph_norm<<<(nNH + EW - 1) / EW, EW, 0, stream>>>(agg1, batch, svar, scnt,
                                                           ngamma, nbeta);
    }

    // 9) final projection -> d_out
    gemm_node_wmma<<<gNode, blk, 0, stream>>>(agg1, HID, final_w, HID, final_b,
                                              nullptr, out, HID, 0, 0);

    (void)in_sizes; (void)n_in; (void)out_size; (void)ws_size;
}